// GCN_MC_39247411151090
// MI455X (gfx1250) — compile-verified
//
#include <hip/hip_runtime.h>

typedef __attribute__((ext_vector_type(2))) float v2f;
typedef __attribute__((ext_vector_type(8))) float v8f;

#define N_NODES 50000
#define N_EDGES 800000
#define D       64   // D_IN == D_OUT == 64

// ---------------------------------------------------------------------------
// Kernel 1: zero the aggregation buffer (12.8 MB in d_ws), float4 stores.
// ---------------------------------------------------------------------------
__global__ void gcn_zero_kernel(float4* __restrict__ p, int n4) {
    int i = blockIdx.x * blockDim.x + threadIdx.x;
    if (i < n4) p[i] = make_float4(0.f, 0.f, 0.f, 0.f);
}

// ---------------------------------------------------------------------------
// Kernel 2: edge scatter-add. One wave (32 lanes) per edge; each lane handles
// 2 consecutive columns -> 64 columns per edge. src/dst loads are
// wave-uniform addresses; atomics are global_atomic_add_f32.
// ---------------------------------------------------------------------------
__global__ void gcn_scatter_kernel(const float* __restrict__ x,
                                   const int* __restrict__ src,
                                   const int* __restrict__ dst,
                                   float* __restrict__ agg) {
    int t = blockIdx.x * blockDim.x + threadIdx.x;
    int e = t >> 5;                 // edge index (one wave32 per edge)
    if (e >= N_EDGES) return;
    int lane = t & 31;
    int s = src[e];
    int d = dst[e];
    int c = lane * 2;
    float2 v = *(const float2*)(x + (size_t)s * D + c);
    atomicAdd(&agg[(size_t)d * D + c],     v.x);
    atomicAdd(&agg[(size_t)d * D + c + 1], v.y);
}

// ---------------------------------------------------------------------------
// Kernel 3: out = relu(agg @ W^T) + x via V_WMMA_F32_16X16X4_F32.
// 12500 16x16 output tiles (3125 M-tiles x 4 N-tiles, no remainders).
// One tile per wave, 8 waves per 256-thread block. K=64 -> 16 chained WMMAs.
// Tile selection is wave-uniform, so EXEC is all-1s at every WMMA.
// ---------------------------------------------------------------------------
__global__ __launch_bounds__(256)
void gcn_wmma_kernel(const float* __restrict__ agg,
                     const float* __restrict__ W,   // [D_OUT][D_IN] row-major
                     const float* __restrict__ x,
                     float* __restrict__ out) {
    const int lane = threadIdx.x & 31;
    const int wave = threadIdx.x >> 5;
    const int tile = blockIdx.x * 8 + wave;
    const int n_tiles = (N_NODES / 16) * (D / 16);   // 12500
    if (tile >= n_tiles) return;                     // wave-uniform exit

    const int mt   = tile >> 2;      // M tile (0..3124)
    const int nt   = tile & 3;       // N tile (0..3)
    const int half = lane >> 4;      // lane-half: selects K pair {0,1} vs {2,3}
    const int l15  = lane & 15;
    const int khi  = half * 2;

    // A fragment source: agg row (M = l15 within tile), K offset khi
    const float* aptr = agg + ((size_t)(mt * 16 + l15)) * D + khi;
    // B fragment source: B[k][n] = W[n][k]  -> row n of W, K offset khi
    const float* bptr = W + ((size_t)(nt * 16 + l15)) * D + khi;

    v8f c = {};
#pragma unroll
    for (int kk = 0; kk < D; kk += 4) {
        v2f a, b;
        a.x = aptr[kk];     a.y = aptr[kk + 1];
        b.x = bptr[kk];     b.y = bptr[kk + 1];
        // 8 args: (neg_a, A, neg_b, B, c_mod, C, reuse_a, reuse_b)
        c = __builtin_amdgcn_wmma_f32_16x16x4_f32(
                false, a, false, b, (short)0, c, false, false);
    }

    // C/D layout: VGPR v holds row v (lanes 0-15) or v+8 (lanes 16-31),
    // column = lane & 15. Fuse relu + residual on writeback.
    const int rbase = mt * 16 + half * 8;
    const int col   = nt * 16 + l15;
#pragma unroll
    for (int v = 0; v < 8; ++v) {
        const size_t idx = (size_t)(rbase + v) * D + col;
        float h = c[v];
        h = h > 0.0f ? h : 0.0f;
        out[idx] = h + x[idx];
    }
}

// ---------------------------------------------------------------------------
extern "C" void kernel_launch(void* const* d_in, const int* in_sizes, int n_in,
                              void* d_out, int out_size, void* d_ws, size_t ws_size,
                              hipStream_t stream) {
    const float* x   = (const float*)d_in[0];   // [N_NODES, D]
    const float* W   = (const float*)d_in[1];   // [D, D]
    const int*   src = (const int*)d_in[2];     // [N_EDGES]
    const int*   dst = (const int*)d_in[3];     // [N_EDGES]
    float*       out = (float*)d_out;           // [N_NODES, D]
    float*       agg = (float*)d_ws;            // scratch: N_NODES*D floats

    // 1) zero agg
    const int n4 = N_NODES * D / 4;
    gcn_zero_kernel<<<(n4 + 255) / 256, 256, 0, stream>>>((float4*)agg, n4);

    // 2) scatter-add edges
    const long long total = (long long)N_EDGES * 32;
    gcn_scatter_kernel<<<(int)((total + 255) / 256), 256, 0, stream>>>(x, src, dst, agg);

    // 3) WMMA GEMM + relu + residual
    const int n_tiles = (N_NODES / 16) * (D / 16);  // 12500
    gcn_wmma_kernel<<<(n_tiles + 7) / 8, 256, 0, stream>>>(agg, W, x, out);
}